// TripletLoss_63170378989855
// MI455X (gfx1250) — compile-verified
//
#include <hip/hip_runtime.h>
#include <hip/hip_bf16.h>

typedef float v2f __attribute__((ext_vector_type(2)));
typedef float v8f __attribute__((ext_vector_type(8)));

#define NROWS 1024
#define DIM   512
#define MARGIN 1.0f
#define MAXPOS 160   // binomial(1024, 1/32): mean 32, max far below 160

// ---------------------------------------------------------------------------
// Kernel 1: sq[i] = ||x_i||^2.  One wave32 per row, coalesced strided loads,
// wave shuffle reduction.
// ---------------------------------------------------------------------------
__global__ void sq_kernel(const float* __restrict__ F, float* __restrict__ sq) {
    int wave = threadIdx.x >> 5;
    int lane = threadIdx.x & 31;
    int row  = blockIdx.x * 8 + wave;
    const float* r = F + row * DIM;
    float s = 0.f;
    for (int k = lane; k < DIM; k += 32) { float v = r[k]; s += v * v; }
    for (int off = 16; off > 0; off >>= 1) s += __shfl_xor(s, off, 32);
    if (lane == 0) sq[row] = s;
}

__device__ __forceinline__ float dist_from_d2(float d2) {
    d2 = fmaxf(d2, 0.0f);
    return (d2 > 0.0f) ? sqrtf(d2) : 0.0f;   // zero-distance -> exactly 0
}

// ---------------------------------------------------------------------------
// Kernel 2: G = X * X^T via V_WMMA_F32_16X16X4_F32 with 2x2 register tiling:
// each wave owns a 32x32 output block (four 16x16 WMMA tiles). Per k-step it
// loads a0,a1,b0,b1 (4 x float2 per lane) and issues 4 WMMAs -> 1 load per
// WMMA instead of 2, with A/B operands reused across the 2x2 block.
// Fused epilogue: dist[m][n] = sqrt(max(sq[m] + sq[n] - 2*G[m][n], 0)).
//
// A 16x4 f32 layout : lane l -> M = l&15, K = 2*(l>>4) + {vgpr0, vgpr1}
// B  4x16 f32 layout: lane l -> N = l&15, K = 2*(l>>4) + {vgpr0, vgpr1}
//   and B[k][n] = X[n][k]  =>  both A and B are contiguous float2 row loads.
// C/D 16x16 f32: vgpr r, lane l -> M = r + 8*(l>>4), N = l&15.
// Branch-free: EXEC stays all-ones as WMMA requires.
// ---------------------------------------------------------------------------
__global__ void gemm_dist_kernel(const float* __restrict__ F,
                                 const float* __restrict__ sq,
                                 float* __restrict__ dist) {
    int lane = threadIdx.x & 31;
    int tile = blockIdx.x * 4 + (threadIdx.x >> 5);   // 1024 supertiles (32x32)
    int ti = tile >> 5;                                // 32x32 supertile grid
    int tj = tile & 31;
    int m16 = lane & 15;
    int hi  = lane >> 4;
    int koff = 2 * hi;

    const float* a0p = F + (ti * 32 +      m16) * DIM + koff;
    const float* a1p = F + (ti * 32 + 16 + m16) * DIM + koff;
    const float* b0p = F + (tj * 32 +      m16) * DIM + koff;
    const float* b1p = F + (tj * 32 + 16 + m16) * DIM + koff;

    v8f c00 = {0.f,0.f,0.f,0.f,0.f,0.f,0.f,0.f};
    v8f c01 = c00, c10 = c00, c11 = c00;

#pragma unroll 4
    for (int k = 0; k < DIM; k += 4) {
        v2f a0 = *(const v2f*)(a0p + k);
        v2f a1 = *(const v2f*)(a1p + k);
        v2f b0 = *(const v2f*)(b0p + k);
        v2f b1 = *(const v2f*)(b1p + k);
        // (neg_a, A, neg_b, B, c_mod, C, reuse_a, reuse_b)
        c00 = __builtin_amdgcn_wmma_f32_16x16x4_f32(false, a0, false, b0,
                                                    (short)0, c00, false, false);
        c01 = __builtin_amdgcn_wmma_f32_16x16x4_f32(false, a0, false, b1,
                                                    (short)0, c01, false, false);
        c10 = __builtin_amdgcn_wmma_f32_16x16x4_f32(false, a1, false, b0,
                                                    (short)0, c10, false, false);
        c11 = __builtin_amdgcn_wmma_f32_16x16x4_f32(false, a1, false, b1,
                                                    (short)0, c11, false, false);
    }

    int n0 = tj * 32 + m16;
    int n1 = n0 + 16;
    float sqn0 = sq[n0];
    float sqn1 = sq[n1];
#pragma unroll
    for (int r = 0; r < 8; ++r) {
        int m0 = ti * 32 + r + 8 * hi;
        int m1 = m0 + 16;
        float sqm0 = sq[m0];
        float sqm1 = sq[m1];
        dist[m0 * NROWS + n0] = dist_from_d2(sqm0 + sqn0 - 2.0f * c00[r]);
        dist[m0 * NROWS + n1] = dist_from_d2(sqm0 + sqn1 - 2.0f * c01[r]);
        dist[m1 * NROWS + n0] = dist_from_d2(sqm1 + sqn0 - 2.0f * c10[r]);
        dist[m1 * NROWS + n1] = dist_from_d2(sqm1 + sqn1 - 2.0f * c11[r]);
    }
}

// ---------------------------------------------------------------------------
// Kernel 3: per-anchor loss. Block = 256 threads per anchor.
// Stage dist row + labels in LDS, compact positive distances (includes the
// diagonal d=0, as the reference's pos mask does) in deterministic order,
// then each thread accumulates over its strided negatives.
// ---------------------------------------------------------------------------
__global__ void loss_anchor_kernel(const float* __restrict__ dist,
                                   const int* __restrict__ labels,
                                   float* __restrict__ partial) {
    __shared__ float row[NROWS];     // 4 KB
    __shared__ int   lab[NROWS];     // 4 KB
    __shared__ float posd[MAXPOS];
    __shared__ int   npos_s;
    __shared__ float red[256];

    int i   = blockIdx.x;
    int tid = threadIdx.x;

    for (int j = tid; j < NROWS; j += 256) {
        row[j] = dist[i * NROWS + j];
        lab[j] = labels[j];
    }
    __syncthreads();

    int li = lab[i];
    if (tid == 0) {                  // deterministic compaction order
        int np = 0;
        for (int j = 0; j < NROWS; ++j)
            if (lab[j] == li && np < MAXPOS) posd[np++] = row[j];
        npos_s = np;
    }
    __syncthreads();

    int np = npos_s;
    float local = 0.f;
    for (int j = tid; j < NROWS; j += 256) {
        if (lab[j] != li) {
            float base = MARGIN - row[j];
            for (int p = 0; p < np; ++p)
                local += fmaxf(posd[p] + base, 0.f);
        }
    }
    red[tid] = local;
    __syncthreads();
    for (int s = 128; s > 0; s >>= 1) {
        if (tid < s) red[tid] += red[tid + s];
        __syncthreads();
    }
    if (tid == 0) partial[i] = red[0];
}

// ---------------------------------------------------------------------------
// Kernel 4: deterministic final reduction of the 1024 per-anchor sums.
// ---------------------------------------------------------------------------
__global__ void final_reduce_kernel(const float* __restrict__ partial,
                                    float* __restrict__ out, float inv_n) {
    __shared__ float red[256];
    int tid = threadIdx.x;
    float s = 0.f;
    for (int j = tid; j < NROWS; j += 256) s += partial[j];
    red[tid] = s;
    __syncthreads();
    for (int st = 128; st > 0; st >>= 1) {
        if (tid < st) red[tid] += red[tid + st];
        __syncthreads();
    }
    if (tid == 0) out[0] = red[0] * inv_n;
}

extern "C" void kernel_launch(void* const* d_in, const int* in_sizes, int n_in,
                              void* d_out, int out_size, void* d_ws, size_t ws_size,
                              hipStream_t stream) {
    const float* F      = (const float*)d_in[0];   // [1024, 512] fp32
    const int*   labels = (const int*)d_in[1];     // [1024] int32
    float*       out    = (float*)d_out;           // scalar fp32

    char*  ws   = (char*)d_ws;
    float* dist = (float*)ws;                                // 4 MB
    float* sq   = (float*)(ws + (size_t)NROWS * NROWS * 4);  // 4 KB
    float* part = (float*)(ws + (size_t)NROWS * NROWS * 4 + 4096); // 4 KB

    sq_kernel<<<NROWS / 8, 256, 0, stream>>>(F, sq);
    // 1024 supertiles (32x32 each), 4 waves per 128-thread block
    gemm_dist_kernel<<<1024 / 4, 128, 0, stream>>>(F, sq, dist);
    loss_anchor_kernel<<<NROWS, 256, 0, stream>>>(dist, labels, part);

    const float inv_n = (float)(1.0 / (1024.0 + 1e-8));
    final_reduce_kernel<<<1, 256, 0, stream>>>(part, out, inv_n);
}